// Invariants_49409303773207
// MI455X (gfx1250) — compile-verified
//
#include <hip/hip_runtime.h>
#include <cstdint>

#define TPB    256
#define ITEMS  256
#define IN_DW  (ITEMS * 9)   // 2304 floats staged in  per block (9216 B)
#define OUT_DW (ITEMS * 5)   // 1280 floats staged out per block (5120 B)

#if defined(__AMDGCN__) && \
    __has_builtin(__builtin_amdgcn_global_load_async_to_lds_b128) && \
    __has_builtin(__builtin_amdgcn_global_store_async_from_lds_b128)
#define ASYNC_LDS 1
#else
#define ASYNC_LDS 0
#endif

#if ASYNC_LDS
typedef int v4i __attribute__((ext_vector_type(4)));
typedef __attribute__((address_space(1))) v4i* global_v4i_p;
typedef __attribute__((address_space(3))) v4i* lds_v4i_p;

__device__ __forceinline__ global_v4i_p as_global_v4(const void* p) {
  // generic pointer to global memory has identical bits to an AS1 pointer
  return (global_v4i_p)(uintptr_t)p;
}
__device__ __forceinline__ lds_v4i_p as_lds_v4(void* p) {
  // flat LDS address: low 32 bits are the LDS byte offset (aperture in high bits)
  return (lds_v4i_p)(uint32_t)(uintptr_t)p;
}
__device__ __forceinline__ void wait_async0() {
#if __has_builtin(__builtin_amdgcn_s_wait_asynccnt)
  __builtin_amdgcn_s_wait_asynccnt(0);
#else
  asm volatile("s_wait_asynccnt 0" ::: "memory");
#endif
}
#endif

__global__ __launch_bounds__(TPB) void Invariants_49409303773207_kernel(
    const float* __restrict__ Fm,   // [n,3,3] row-major
    const float* __restrict__ Gm,   // [3,3]
    float* __restrict__ out,        // [n,5]: I1, J, -J, I4, I5
    int n)
{
#if ASYNC_LDS
  __shared__ float sIn[IN_DW];
  __shared__ float sOut[OUT_DW];
#endif
  const int  t    = (int)threadIdx.x;
  const long long base = (long long)blockIdx.x * ITEMS;
  const int  item = (int)base + t;
#if ASYNC_LDS
  const bool fullTile = (base + ITEMS) <= (long long)n;   // uniform per block
#endif

  // G is uniform across the grid -> scalar loads
  const float g00 = Gm[0], g01 = Gm[1], g02 = Gm[2];
  const float g10 = Gm[3], g11 = Gm[4], g12 = Gm[5];
  const float g20 = Gm[6], g21 = Gm[7], g22 = Gm[8];

  float f0, f1, f2, f3, f4, f5, f6, f7, f8;

#if ASYNC_LDS
  if (fullTile) {
    // Coalesced async stage-in: 576 b128 chunks for this block's 256 items.
    float* gsrc = const_cast<float*>(Fm) + base * 9;
    for (int q = t; q < IN_DW / 4; q += TPB) {
      __builtin_amdgcn_global_load_async_to_lds_b128(
          as_global_v4(gsrc + q * 4), as_lds_v4(&sIn[q * 4]), 0, 0);
    }
    wait_async0();
    __syncthreads();
    const int o = t * 9;   // stride 9 dwords across lanes -> conflict-free (gcd(9,64)=1)
    f0 = sIn[o + 0]; f1 = sIn[o + 1]; f2 = sIn[o + 2];
    f3 = sIn[o + 3]; f4 = sIn[o + 4]; f5 = sIn[o + 5];
    f6 = sIn[o + 6]; f7 = sIn[o + 7]; f8 = sIn[o + 8];
  } else
#endif
  {
    if (item < n) {
      const float* p = Fm + (long long)item * 9;
      f0 = p[0]; f1 = p[1]; f2 = p[2];
      f3 = p[3]; f4 = p[4]; f5 = p[5];
      f6 = p[6]; f7 = p[7]; f8 = p[8];
    } else {
      f0 = 1.f; f1 = 0.f; f2 = 0.f;
      f3 = 0.f; f4 = 1.f; f5 = 0.f;
      f6 = 0.f; f7 = 0.f; f8 = 1.f;
    }
  }

  // C = F^T F (symmetric): C[j][k] = sum_i F[i][j]*F[i][k]
  const float C00 = f0 * f0 + f3 * f3 + f6 * f6;
  const float C01 = f0 * f1 + f3 * f4 + f6 * f7;
  const float C02 = f0 * f2 + f3 * f5 + f6 * f8;
  const float C11 = f1 * f1 + f4 * f4 + f7 * f7;
  const float C12 = f1 * f2 + f4 * f5 + f7 * f8;
  const float C22 = f2 * f2 + f5 * f5 + f8 * f8;

  // J = det(F)
  const float J = f0 * (f4 * f8 - f5 * f7)
                - f1 * (f3 * f8 - f5 * f6)
                + f2 * (f3 * f7 - f4 * f6);

  const float I1 = C00 + C11 + C22;

  // cof(C) = detC * inv(C) = adj(C); symmetric since C is symmetric
  const float k00 = C11 * C22 - C12 * C12;
  const float k11 = C00 * C22 - C02 * C02;
  const float k22 = C00 * C11 - C01 * C01;
  const float k01 = C02 * C12 - C01 * C22;
  const float k02 = C01 * C12 - C11 * C02;
  const float k12 = C01 * C02 - C00 * C12;

  // I4 = sum_ij C[i][j]*G[j][i];  I5 = sum_ij cofC[i][j]*G[j][i]
  const float s01 = g01 + g10, s02 = g02 + g20, s12 = g12 + g21;
  const float I4 = C00 * g00 + C11 * g11 + C22 * g22
                 + C01 * s01 + C02 * s02 + C12 * s12;
  const float I5 = k00 * g00 + k11 * g11 + k22 * g22
                 + k01 * s01 + k02 * s02 + k12 * s12;

  const float r0 = I1, r1 = J, r2 = -J, r3 = I4, r4 = I5;

#if ASYNC_LDS
  if (fullTile) {
    const int so = t * 5;
    sOut[so + 0] = r0; sOut[so + 1] = r1; sOut[so + 2] = r2;
    sOut[so + 3] = r3; sOut[so + 4] = r4;
    __syncthreads();
    // Coalesced async stage-out: 320 b128 chunks; S_ENDPGM's implicit
    // wait-idle guarantees ASYNCcnt drains before the wave retires.
    float* gdst = out + base * 5;
    for (int q = t; q < OUT_DW / 4; q += TPB) {
      __builtin_amdgcn_global_store_async_from_lds_b128(
          as_global_v4(gdst + q * 4), as_lds_v4(&sOut[q * 4]), 0, 0);
    }
  } else
#endif
  {
    if (item < n) {
      float* p = out + (long long)item * 5;
      p[0] = r0; p[1] = r1; p[2] = r2; p[3] = r3; p[4] = r4;
    }
  }
}

extern "C" void kernel_launch(void* const* d_in, const int* in_sizes, int n_in,
                              void* d_out, int out_size, void* d_ws, size_t ws_size,
                              hipStream_t stream) {
  (void)n_in; (void)out_size; (void)d_ws; (void)ws_size;
  const float* F = (const float*)d_in[0];
  const float* G = (const float*)d_in[1];
  float* out = (float*)d_out;
  const int n = in_sizes[0] / 9;
  const int blocks = (n + ITEMS - 1) / ITEMS;
  Invariants_49409303773207_kernel<<<blocks, TPB, 0, stream>>>(F, G, out, n);
}